// scGAT_PReLU_40106404610256
// MI455X (gfx1250) — compile-verified
//
#include <hip/hip_runtime.h>

typedef __attribute__((ext_vector_type(2))) float v2f;
typedef __attribute__((ext_vector_type(8))) float v8f;
typedef __attribute__((ext_vector_type(4))) unsigned int v4u;
typedef __attribute__((ext_vector_type(8))) int v8i;
typedef __attribute__((ext_vector_type(4))) int v4i;

#if defined(__has_builtin)
#if __has_builtin(__builtin_amdgcn_tensor_load_to_lds) && \
    __has_builtin(__builtin_amdgcn_s_wait_tensorcnt)
#define GAT_HAVE_TDM 1
#endif
#endif
#ifndef GAT_HAVE_TDM
#define GAT_HAVE_TDM 0
#endif

#if GAT_HAVE_TDM
// Issue one TDM 2D tile load: tileW x tileH elements of f32, row stride
// `strideElems`, from global `gptr` into LDS byte offset `ldsOff`.
// Wave-uniform args; call from a single wave, then s_wait_tensorcnt(0).
__device__ __forceinline__ void gat_tdm_load_tile_f32(
    const float* gptr, unsigned ldsOff, unsigned tileW, unsigned tileH,
    unsigned strideElems, unsigned remW, unsigned remH) {
    unsigned long long ga = (unsigned long long)(const void*)gptr;
    v4u g0;
    g0[0] = 1u;                                        // count=1 (valid), user mode
    g0[1] = ldsOff;                                    // lds_addr (bytes)
    g0[2] = (unsigned)(ga & 0xffffffffull);            // global_addr[31:0]
    g0[3] = (unsigned)((ga >> 32) & 0x01ffffffull)     // global_addr[56:32]
            | (2u << 30);                              // type=2 ("image")
    v8i g1;
    g1[0] = (int)(2u << 16);                           // data_size=2 -> 4 bytes
    g1[1] = (int)((remW & 0xffffu) << 16);             // tensor_dim0[15:0]
    g1[2] = (int)(((remW >> 16) & 0xffffu) | ((remH & 0xffffu) << 16));
    g1[3] = (int)(((remH >> 16) & 0xffffu) | (tileW << 16));   // tile_dim0
    g1[4] = (int)tileH;                                // tile_dim1 (tile_dim2=0)
    g1[5] = (int)strideElems;                          // tensor_dim0_stride lo
    g1[6] = 0;
    g1[7] = 0;
    v4i gz4 = {0, 0, 0, 0};
    v8i gz8 = {0, 0, 0, 0, 0, 0, 0, 0};
    __builtin_amdgcn_tensor_load_to_lds(g0, g1, gz4, gz4, gz8, 0);
}
#endif

// ---------------------------------------------------------------------------
// GEMM1: H1[N,64] = X[N,512] @ W1[512,64]   (f32 WMMA 16x16x4, 4 waves/block)
// A tiles staged to LDS by the Tensor Data Mover (TENSORcnt), B from global.
// ---------------------------------------------------------------------------
__global__ void __launch_bounds__(128) gat_gemm1_wmma(const float* __restrict__ X,
                                                      const float* __restrict__ W,
                                                      float* __restrict__ Hout,
                                                      int Nn) {
    __shared__ float As[16][64];
    const int tid   = threadIdx.x;
    const int wave  = tid >> 5;        // 0..3 -> 16-col tile
    const int lane  = tid & 31;
    const int lhalf = lane >> 4;       // 0 or 1 (selects K pair / M half)
    const int l16   = lane & 15;
    const int row0  = blockIdx.x * 16;
    const int col   = wave * 16 + l16; // 0..63

    v8f acc = {};
    for (int k0 = 0; k0 < 512; k0 += 64) {
#if GAT_HAVE_TDM
        if (wave == 0) {
            unsigned ldsOff = (unsigned)(unsigned long long)&As[0][0];
            gat_tdm_load_tile_f32(X + (long)row0 * 512 + k0, ldsOff,
                                  /*tileW=*/64, /*tileH=*/16,
                                  /*stride=*/512,
                                  /*remW=*/(unsigned)(512 - k0),
                                  /*remH=*/(unsigned)(Nn - row0));
            __builtin_amdgcn_s_wait_tensorcnt(0);
        }
        __syncthreads();
#else
        // cooperative, coalesced A-tile load (16 rows x 64 K), clamp rows
        for (int i = 0; i < 8; ++i) {
            int idx = tid + i * 128;       // 0..1023
            int r   = idx >> 6;
            int kk  = idx & 63;
            int gr  = row0 + r; if (gr >= Nn) gr = Nn - 1;
            As[r][kk] = X[(long)gr * 512 + k0 + kk];
        }
        __syncthreads();
#endif
        for (int kk = 0; kk < 64; kk += 4) {
            v2f a, b;
            // A 16x4 f32 fragment: lanes 0-15 K={0,1}, lanes 16-31 K={2,3}
            a.x = As[l16][kk + 2 * lhalf];
            a.y = As[l16][kk + 2 * lhalf + 1];
            // B 4x16 f32 fragment: mirrored layout, N = lane%16
            int krow = k0 + kk + 2 * lhalf;
            b.x = W[krow * 64 + col];
            b.y = W[(krow + 1) * 64 + col];
            acc = __builtin_amdgcn_wmma_f32_16x16x4_f32(
                false, a, false, b, (short)0, acc, false, false);
        }
        __syncthreads();
    }
    // C/D layout: VGPR g -> M = g + 8*lhalf, N = lane%16
    for (int g = 0; g < 8; ++g) {
        int r = row0 + g + 8 * lhalf;
        if (r < Nn) Hout[(long)r * 64 + col] = acc[g];
    }
}

// ---------------------------------------------------------------------------
// GEMM2: H2[N,128] = G[N,64] @ W2[64,128]   (8 waves/block, single K chunk)
// ---------------------------------------------------------------------------
__global__ void __launch_bounds__(256) gat_gemm2_wmma(const float* __restrict__ G,
                                                      const float* __restrict__ W,
                                                      float* __restrict__ Hout,
                                                      int Nn) {
    __shared__ float As[16][64];
    const int tid   = threadIdx.x;
    const int wave  = tid >> 5;         // 0..7
    const int lane  = tid & 31;
    const int lhalf = lane >> 4;
    const int l16   = lane & 15;
    const int row0  = blockIdx.x * 16;
    const int col   = wave * 16 + l16;  // 0..127

#if GAT_HAVE_TDM
    if (wave == 0) {
        unsigned ldsOff = (unsigned)(unsigned long long)&As[0][0];
        gat_tdm_load_tile_f32(G + (long)row0 * 64, ldsOff,
                              /*tileW=*/64, /*tileH=*/16,
                              /*stride=*/64,
                              /*remW=*/64u,
                              /*remH=*/(unsigned)(Nn - row0));
        __builtin_amdgcn_s_wait_tensorcnt(0);
    }
    __syncthreads();
#else
    for (int i = 0; i < 4; ++i) {
        int idx = tid + i * 256;        // 0..1023
        int r   = idx >> 6;
        int kk  = idx & 63;
        int gr  = row0 + r; if (gr >= Nn) gr = Nn - 1;
        As[r][kk] = G[(long)gr * 64 + kk];
    }
    __syncthreads();
#endif

    v8f acc = {};
    for (int kk = 0; kk < 64; kk += 4) {
        v2f a, b;
        a.x = As[l16][kk + 2 * lhalf];
        a.y = As[l16][kk + 2 * lhalf + 1];
        int krow = kk + 2 * lhalf;
        b.x = W[krow * 128 + col];
        b.y = W[(krow + 1) * 128 + col];
        acc = __builtin_amdgcn_wmma_f32_16x16x4_f32(
            false, a, false, b, (short)0, acc, false, false);
    }
    for (int g = 0; g < 8; ++g) {
        int r = row0 + g + 8 * lhalf;
        if (r < Nn) Hout[(long)r * 128 + col] = acc[g];
    }
}

// ---------------------------------------------------------------------------
// Per-node attention coefficients: as[n,h] = sum_c h[n,h,c]*a_src[h,c], etc.
// ---------------------------------------------------------------------------
__global__ void gat_alpha_kernel(const float* __restrict__ Hf,
                                 const float* __restrict__ a_src,
                                 const float* __restrict__ a_dst,
                                 float* __restrict__ as_, float* __restrict__ ad_,
                                 int Nn, int Hh, int C) {
    long i = (long)blockIdx.x * blockDim.x + threadIdx.x;
    if (i >= (long)Nn * Hh) return;
    int n = (int)(i / Hh), h = (int)(i % Hh);
    const float* hp = Hf + (long)n * Hh * C + h * C;
    float s = 0.f, d = 0.f;
    for (int c = 0; c < C; ++c) {
        float v = hp[c];
        s += v * a_src[h * C + c];
        d += v * a_dst[h * C + c];
    }
    as_[i] = s;
    ad_[i] = d;
}

__global__ void gat_fill_kernel(float* __restrict__ p, float v, long n) {
    long i = (long)blockIdx.x * blockDim.x + threadIdx.x;
    if (i < n) p[i] = v;
}

__device__ __forceinline__ void edge_sd(const int* __restrict__ ei, long e, long E,
                                        int& src, int& dst) {
    if (e < E) { src = ei[e]; dst = ei[E + e]; }
    else       { int v = (int)(e - E); src = v; dst = v; }   // self-loop
}

__device__ __forceinline__ float leaky02(float s) {
    return s >= 0.f ? s : 0.2f * s;
}

__device__ __forceinline__ void atomicMaxF32(float* addr, float val) {
    int* ai = (int*)addr;
    int cur = __float_as_int(*(volatile float*)addr);
    while (val > __int_as_float(cur)) {
        int prev = atomicCAS(ai, cur, __float_as_int(val));
        if (prev == cur) break;
        cur = prev;
    }
}

// Pass 1: segment max of leaky_relu scores, grouped by dst
__global__ void gat_edge_max_kernel(const int* __restrict__ ei, long E, int Nn,
                                    const float* __restrict__ as_,
                                    const float* __restrict__ ad_,
                                    float* __restrict__ m, int Hh) {
    long e = (long)blockIdx.x * blockDim.x + threadIdx.x;
    if (e >= E + Nn) return;
    int src, dst; edge_sd(ei, e, E, src, dst);
    for (int h = 0; h < Hh; ++h) {
        float lr = leaky02(as_[(long)src * Hh + h] + ad_[(long)dst * Hh + h]);
        atomicMaxF32(&m[(long)dst * Hh + h], lr);
    }
}

// Pass 2: segment sum of exp(score - max)
__global__ void gat_edge_denom_kernel(const int* __restrict__ ei, long E, int Nn,
                                      const float* __restrict__ as_,
                                      const float* __restrict__ ad_,
                                      const float* __restrict__ m,
                                      float* __restrict__ den, int Hh) {
    long e = (long)blockIdx.x * blockDim.x + threadIdx.x;
    if (e >= E + Nn) return;
    int src, dst; edge_sd(ei, e, E, src, dst);
    for (int h = 0; h < Hh; ++h) {
        float lr = leaky02(as_[(long)src * Hh + h] + ad_[(long)dst * Hh + h]);
        float ex = expf(lr - m[(long)dst * Hh + h]);
        atomicAdd(&den[(long)dst * Hh + h], ex);
    }
}

// Pass 3 layer-1: out[dst, h, c] += alpha * h[src, h, c]   (H=8, C=8)
__global__ void gat_edge_agg1_kernel(const int* __restrict__ ei, long E, int Nn,
                                     const float* __restrict__ as_,
                                     const float* __restrict__ ad_,
                                     const float* __restrict__ m,
                                     const float* __restrict__ den,
                                     const float* __restrict__ feat,
                                     float* __restrict__ outp) {
    long e = (long)blockIdx.x * blockDim.x + threadIdx.x;
    if (e >= E + Nn) return;
    int src, dst; edge_sd(ei, e, E, src, dst);
    const float* fsrc = feat + (long)src * 64;
    __builtin_prefetch(fsrc, 0, 0);
    float* od = outp + (long)dst * 64;
    for (int h = 0; h < 8; ++h) {
        float lr    = leaky02(as_[(long)src * 8 + h] + ad_[(long)dst * 8 + h]);
        float alpha = expf(lr - m[(long)dst * 8 + h]) / den[(long)dst * 8 + h];
        const float* fh = fsrc + h * 8;
        float* oh       = od + h * 8;
        for (int c = 0; c < 8; ++c) atomicAdd(&oh[c], alpha * fh[c]);
    }
}

// Pass 3 layer-2: heads folded in registers (mean later): o2[dst,c] += sum_h alpha_h*h2[src,h,c]
__global__ void gat_edge_agg2_kernel(const int* __restrict__ ei, long E, int Nn,
                                     const float* __restrict__ as_,
                                     const float* __restrict__ ad_,
                                     const float* __restrict__ m,
                                     const float* __restrict__ den,
                                     const float* __restrict__ feat,
                                     float* __restrict__ outp) {
    long e = (long)blockIdx.x * blockDim.x + threadIdx.x;
    if (e >= E + Nn) return;
    int src, dst; edge_sd(ei, e, E, src, dst);
    const float* fsrc = feat + (long)src * 128;
    __builtin_prefetch(fsrc, 0, 0);
    float acc[16];
    for (int c = 0; c < 16; ++c) acc[c] = 0.f;
    for (int h = 0; h < 8; ++h) {
        float lr    = leaky02(as_[(long)src * 8 + h] + ad_[(long)dst * 8 + h]);
        float alpha = expf(lr - m[(long)dst * 8 + h]) / den[(long)dst * 8 + h];
        const float* fh = fsrc + h * 16;
        for (int c = 0; c < 16; ++c) acc[c] += alpha * fh[c];
    }
    float* od = outp + (long)dst * 16;
    for (int c = 0; c < 16; ++c) atomicAdd(&od[c], acc[c]);
}

// g = prelu(g + bias1), in place over [N,64]
__global__ void gat_finalize1_kernel(float* __restrict__ g,
                                     const float* __restrict__ bias,
                                     const float* __restrict__ prelu_w, long n64) {
    long i = (long)blockIdx.x * blockDim.x + threadIdx.x;
    if (i >= n64) return;
    float v = g[i] + bias[i & 63];
    float w = *prelu_w;
    g[i] = v >= 0.f ? v : w * v;
}

// out = log_softmax(o2/H + bias2) over 16 classes
__global__ void gat_finalize2_kernel(const float* __restrict__ o2,
                                     const float* __restrict__ bias2,
                                     float* __restrict__ out, int Nn) {
    int n = blockIdx.x * blockDim.x + threadIdx.x;
    if (n >= Nn) return;
    float v[16];
    float mx = -__builtin_inff();
    for (int c = 0; c < 16; ++c) {
        float t = o2[(long)n * 16 + c] * 0.125f + bias2[c];
        v[c] = t;
        mx = t > mx ? t : mx;
    }
    float ssum = 0.f;
    for (int c = 0; c < 16; ++c) ssum += expf(v[c] - mx);
    float lse = mx + logf(ssum);
    for (int c = 0; c < 16; ++c) out[(long)n * 16 + c] = v[c] - lse;
}

// ---------------------------------------------------------------------------
extern "C" void kernel_launch(void* const* d_in, const int* in_sizes, int n_in,
                              void* d_out, int out_size, void* d_ws, size_t ws_size,
                              hipStream_t stream) {
    const float* x        = (const float*)d_in[0];
    const int*   ei       = (const int*)d_in[1];
    const float* W1       = (const float*)d_in[2];
    const float* att_src1 = (const float*)d_in[3];
    const float* att_dst1 = (const float*)d_in[4];
    const float* bias1    = (const float*)d_in[5];
    const float* prelu_w  = (const float*)d_in[6];
    const float* W2       = (const float*)d_in[7];
    const float* att_src2 = (const float*)d_in[8];
    const float* att_dst2 = (const float*)d_in[9];
    const float* bias2    = (const float*)d_in[10];
    float* out = (float*)d_out;

    const int  Nn = in_sizes[0] / 512;
    const long E  = in_sizes[1] / 2;
    const long ET = E + Nn;

    float* ws = (float*)d_ws;
    size_t off = 0;
    auto carve = [&](size_t n) { float* p = ws + off; off += n; return p; };
    float* h1  = carve((size_t)Nn * 64);
    float* as1 = carve((size_t)Nn * 8);
    float* ad1 = carve((size_t)Nn * 8);
    float* m1  = carve((size_t)Nn * 8);
    float* d1  = carve((size_t)Nn * 8);
    float* g1  = carve((size_t)Nn * 64);
    float* h2  = carve((size_t)Nn * 128);
    float* as2 = carve((size_t)Nn * 8);
    float* ad2 = carve((size_t)Nn * 8);
    float* m2  = carve((size_t)Nn * 8);
    float* d2  = carve((size_t)Nn * 8);
    float* o2  = carve((size_t)Nn * 16);

    const float NEG_INF = -__builtin_inff();
    const unsigned TPB = 256;
    auto gdiv = [](long n, unsigned b) { return (unsigned)((n + b - 1) / b); };

    // ---- Layer 1 ----
    gat_gemm1_wmma<<<gdiv(Nn, 16), 128, 0, stream>>>(x, W1, h1, Nn);
    gat_alpha_kernel<<<gdiv((long)Nn * 8, TPB), TPB, 0, stream>>>(
        h1, att_src1, att_dst1, as1, ad1, Nn, 8, 8);
    gat_fill_kernel<<<gdiv((long)Nn * 8, TPB), TPB, 0, stream>>>(m1, NEG_INF, (long)Nn * 8);
    gat_fill_kernel<<<gdiv((long)Nn * 8, TPB), TPB, 0, stream>>>(d1, 0.f, (long)Nn * 8);
    gat_fill_kernel<<<gdiv((long)Nn * 64, TPB), TPB, 0, stream>>>(g1, 0.f, (long)Nn * 64);
    gat_edge_max_kernel<<<gdiv(ET, TPB), TPB, 0, stream>>>(ei, E, Nn, as1, ad1, m1, 8);
    gat_edge_denom_kernel<<<gdiv(ET, TPB), TPB, 0, stream>>>(ei, E, Nn, as1, ad1, m1, d1, 8);
    gat_edge_agg1_kernel<<<gdiv(ET, TPB), TPB, 0, stream>>>(ei, E, Nn, as1, ad1, m1, d1, h1, g1);
    gat_finalize1_kernel<<<gdiv((long)Nn * 64, TPB), TPB, 0, stream>>>(
        g1, bias1, prelu_w, (long)Nn * 64);

    // ---- Layer 2 ----
    gat_gemm2_wmma<<<gdiv(Nn, 16), 256, 0, stream>>>(g1, W2, h2, Nn);
    gat_alpha_kernel<<<gdiv((long)Nn * 8, TPB), TPB, 0, stream>>>(
        h2, att_src2, att_dst2, as2, ad2, Nn, 8, 16);
    gat_fill_kernel<<<gdiv((long)Nn * 8, TPB), TPB, 0, stream>>>(m2, NEG_INF, (long)Nn * 8);
    gat_fill_kernel<<<gdiv((long)Nn * 8, TPB), TPB, 0, stream>>>(d2, 0.f, (long)Nn * 8);
    gat_fill_kernel<<<gdiv((long)Nn * 16, TPB), TPB, 0, stream>>>(o2, 0.f, (long)Nn * 16);
    gat_edge_max_kernel<<<gdiv(ET, TPB), TPB, 0, stream>>>(ei, E, Nn, as2, ad2, m2, 8);
    gat_edge_denom_kernel<<<gdiv(ET, TPB), TPB, 0, stream>>>(ei, E, Nn, as2, ad2, m2, d2, 8);
    gat_edge_agg2_kernel<<<gdiv(ET, TPB), TPB, 0, stream>>>(ei, E, Nn, as2, ad2, m2, d2, h2, o2);
    gat_finalize2_kernel<<<gdiv(Nn, TPB), TPB, 0, stream>>>(o2, bias2, out, Nn);
}